// DAGERC_fusion_26826365731123
// MI455X (gfx1250) — compile-verified
//
#include <hip/hip_runtime.h>
#include <cstddef>
#include <cstdint>

// ---------------------------------------------------------------------------
// Problem constants (B,N,E,Dh,L) = (64,128,1024,300,2)
// out row layout: [H0 (300) | H1 (300) | H2 (300) | features (1024)] = 1924
// ---------------------------------------------------------------------------
#define BB   64
#define NN   128
#define EE   1024
#define DH   300
#define LL   2
#define ODIM (3*DH + EE)     // 1924
#define ROWS (BB*NN)         // 8192
#define BQ   16              // batches per scan workgroup (== one WMMA M-tile)
#define GPAD 912             // padded gate row (900 -> 912)
#define KC   64              // fc1 K-chunk staged in LDS per TDM transfer

typedef __attribute__((ext_vector_type(2))) float v2f;
typedef __attribute__((ext_vector_type(8))) float v8f;

// ---------------------------------------------------------------------------
// Tensor Data Mover (gfx1250): async 2D tile DMA global -> LDS, zero-fill OOB.
// ---------------------------------------------------------------------------
#if defined(__has_builtin)
# if __has_builtin(__builtin_amdgcn_tensor_load_to_lds) && \
     __has_builtin(__builtin_amdgcn_s_wait_tensorcnt)
#  define HAVE_TDM 1
# endif
#endif
#ifndef HAVE_TDM
# define HAVE_TDM 0
#endif

#if HAVE_TDM
typedef unsigned int tdm_v4u __attribute__((ext_vector_type(4)));
typedef int          tdm_v4i __attribute__((ext_vector_type(4)));
typedef int          tdm_v8i __attribute__((ext_vector_type(8)));

// Load a tile_d1 x tile_d0 (rows x contiguous-cols) f32 tile from a row-major
// tensor (row stride = stride0 elems) into LDS at byte offset lds_off.
// Columns >= tensor_d0 are zero-filled by TDM OOB handling.
__device__ __forceinline__ void tdm_load_2d(unsigned lds_off, const float* gptr,
                                            unsigned tensor_d0, unsigned tensor_d1,
                                            unsigned tile_d0, unsigned tile_d1,
                                            unsigned stride0) {
    unsigned long long ga = (unsigned long long)(uintptr_t)gptr;
    tdm_v4u g0;
    g0.x = 1u;                                        // count=1, user mode
    g0.y = lds_off;                                   // lds_addr [63:32]
    g0.z = (unsigned)ga;                              // global_addr [95:64]
    g0.w = (unsigned)(ga >> 32) | (2u << 30);         // addr hi + type=2 [127:96]
    tdm_v8i g1;
    g1[0] = (int)(2u << 16);                          // data_size = 4B
    g1[1] = (int)((tensor_d0 & 0xffffu) << 16);       // tensor_dim0 lo16
    g1[2] = (int)((tensor_d0 >> 16) | ((tensor_d1 & 0xffffu) << 16));
    g1[3] = (int)((tensor_d1 >> 16) | (tile_d0 << 16));   // tile_dim0
    g1[4] = (int)tile_d1;                             // tile_dim1 (tile_dim2=0)
    g1[5] = (int)stride0;                             // tensor_dim0_stride lo32
    g1[6] = 0;
    g1[7] = 0;
    tdm_v4i z4 = {0, 0, 0, 0};
#if __clang_major__ >= 23
    tdm_v8i z8 = {0, 0, 0, 0, 0, 0, 0, 0};
    __builtin_amdgcn_tensor_load_to_lds(g0, g1, z4, z4, z8, 0);
#else
    __builtin_amdgcn_tensor_load_to_lds(g0, g1, z4, z4, 0);
#endif
}
#endif

// ---------------------------------------------------------------------------
// WMMA helpers: V_WMMA_F32_16X16X4_F32 (wave32).
// A (16x4): lane L -> m=L%16, kh=L/16; {K=2kh, K=2kh+1}
// B (4x16): lane L -> n=L%16, kh=L/16; {row 2kh, row 2kh+1}
// C/D(16x16): vgpr r -> M = r + (L/16)*8, N = L%16
// ---------------------------------------------------------------------------
__device__ __forceinline__ v8f wmma_f32(v2f a, v2f b, v8f c) {
    return __builtin_amdgcn_wmma_f32_16x16x4_f32(
        false, a, false, b, (short)0, c, false, false);
}

__device__ __forceinline__ v2f a_frag_g(const float* A, int row0, int k0, int lda, int lane) {
    int m = lane & 15, kh = lane >> 4;
    return *reinterpret_cast<const v2f*>(A + (size_t)(row0 + m) * lda + k0 + 2 * kh);
}
__device__ __forceinline__ v2f a_frag_lds(const float* As, int k0, int lda, int lane) {
    int m = lane & 15, kh = lane >> 4;
    return *reinterpret_cast<const v2f*>(As + m * lda + k0 + 2 * kh);   // ds_load_b64
}

__device__ __forceinline__ float wave_sum(float v) {
    #pragma unroll
    for (int off = 16; off > 0; off >>= 1) v += __shfl_xor(v, off, 32);
    return v;
}
__device__ __forceinline__ float wave_max(float v) {
    #pragma unroll
    for (int off = 16; off > 0; off >>= 1) v = fmaxf(v, __shfl_xor(v, off, 32));
    return v;
}
__device__ __forceinline__ float sigmoidf(float x) { return 1.f / (1.f + __expf(-x)); }

// ---------------------------------------------------------------------------
// K1: H0 = relu(features @ fc1_w + fc1_b)   (8192 x 1024) @ (1024 x 300)
// grid (64, 19), block 256 = 8 waves, each owning one 16x16 tile (128x16/blk).
// B panel (KC x 16) double-buffered in LDS via TDM (async, overlapped).
// ---------------------------------------------------------------------------
__global__ __launch_bounds__(256) void fc1_kernel(
    const float* __restrict__ F, const float* __restrict__ W,
    const float* __restrict__ bias, float* __restrict__ H0, float* __restrict__ out)
{
    __shared__ float s_b[2][KC * 16];
    const int tid = threadIdx.x;
    const int lane = tid & 31, wave = tid >> 5;
    const int row0 = blockIdx.x * 128 + wave * 16;
    const int n0 = blockIdx.y * 16;
    const unsigned td0 = (unsigned)(DH - n0);   // zero-fill cols past 300 via OOB

#if HAVE_TDM
    if (wave == 0)
        tdm_load_2d((unsigned)(uintptr_t)&s_b[0][0], W + n0, td0, KC, 16, KC, DH);
#else
    for (int t = tid; t < KC * 16; t += 256) {
        int col = n0 + (t & 15);
        s_b[0][t] = (col < DH) ? W[(size_t)(t >> 4) * DH + col] : 0.f;
    }
#endif

    v8f acc = {0.f, 0.f, 0.f, 0.f, 0.f, 0.f, 0.f, 0.f};
    const int n = lane & 15, kh = lane >> 4;

    for (int c = 0; c < EE / KC; ++c) {
        const int nb = (c + 1) & 1;
#if HAVE_TDM
        if (wave == 0) {
            if (c + 1 < EE / KC) {
                tdm_load_2d((unsigned)(uintptr_t)&s_b[nb][0],
                            W + (size_t)(c + 1) * KC * DH + n0, td0, KC, 16, KC, DH);
                __builtin_amdgcn_s_wait_tensorcnt((short)1);   // chunk c landed
            } else {
                __builtin_amdgcn_s_wait_tensorcnt((short)0);
            }
        }
#else
        if (c + 1 < EE / KC) {
            for (int t = tid; t < KC * 16; t += 256) {
                int col = n0 + (t & 15);
                s_b[nb][t] = (col < DH) ? W[(size_t)((c + 1) * KC + (t >> 4)) * DH + col] : 0.f;
            }
        }
#endif
        __syncthreads();
        const float* bs = s_b[c & 1];
        const int k0 = c * KC;
        #pragma unroll 4
        for (int kk = 0; kk < KC; kk += 4) {
            v2f a = a_frag_g(F, row0, k0 + kk, EE, lane);
            v2f b;
            b.x = bs[(kk + 2 * kh) * 16 + n];
            b.y = bs[(kk + 2 * kh + 1) * 16 + n];
            acc = wmma_f32(a, b, acc);
        }
        __syncthreads();
    }

    const int col = n0 + n;
    if (col < DH) {
        const float bv = bias[col];
        #pragma unroll
        for (int r = 0; r < 8; ++r) {
            int row = row0 + r + kh * 8;
            float v = acc[r] + bv;
            v = v > 0.f ? v : 0.f;
            H0[(size_t)row * DH + col]   = v;
            out[(size_t)row * ODIM + col] = v;
        }
    }
}

// K2: copy features into out[:, 900:1924]
__global__ __launch_bounds__(256) void copy_feat_kernel(
    const float* __restrict__ F, float* __restrict__ out)
{
    const size_t total = (size_t)ROWS * EE;
    for (size_t i = (size_t)blockIdx.x * blockDim.x + threadIdx.x; i < total;
         i += (size_t)gridDim.x * blockDim.x) {
        size_t r = i >> 10, c = i & (EE - 1);
        out[r * ODIM + 3 * DH + c] = F[i];
    }
}

// ---------------------------------------------------------------------------
// K3: one DAG layer. 4 WGs x 16 batches; each WG runs the 128-step recurrence
// for its batch slice (batches are independent => no cross-WG sync).
// ---------------------------------------------------------------------------
__global__ __launch_bounds__(512) void dag_scan_kernel(
    const float* __restrict__ Hcur, float* __restrict__ Hnext,
    const float* __restrict__ cwih, const float* __restrict__ cwhh,
    const float* __restrict__ cbih, const float* __restrict__ cbhh,
    const float* __restrict__ pwih, const float* __restrict__ pwhh,
    const float* __restrict__ pbih, const float* __restrict__ pbhh,
    const float* __restrict__ gatw, const float* __restrict__ gatb,
    const int*   __restrict__ adj,
    float* __restrict__ gates_ws, float* __restrict__ out, int outoff)
{
    __shared__ float s_x[BQ][DH];
    __shared__ float s_M[BQ][DH];
    __shared__ float s_w[BQ][NN];
    __shared__ float s_kdot[BQ][NN];

    const int tid = threadIdx.x, lane = tid & 31, wave = tid >> 5;
    const int b0 = blockIdx.x * BQ;
    const float* wq = gatw;
    const float* wk = gatw + DH;
    const float bg = gatb[0];
    float* gws = gates_ws + (size_t)blockIdx.x * (4 * BQ * GPAD);

    for (int i = 0; i < NN; ++i) {
        // ---- phase 0: stage x (zero M at step 0) ----
        for (int t = tid; t < BQ * DH; t += 512) {
            int b = t / DH, d = t - b * DH;
            s_x[b][d] = Hcur[((size_t)(b0 + b) * NN + i) * DH + d];
            if (i == 0) s_M[b][d] = 0.f;
        }
        __syncthreads();

        // ---- phase 1: attention (wave b handles batch b) ----
        if (i > 0) {
            const int b = wave;
            float xq = 0.f;
            for (int d = lane; d < DH; d += 32) xq += s_x[b][d] * wq[d];
            xq = wave_sum(xq);

            const int* arow = adj + ((size_t)(b0 + b) * NN + i) * NN;
            float mx = -3.4e38f;
            for (int j = lane; j < i; j += 32) {
                float aj = (arow[j] != 0) ? (xq + s_kdot[b][j] + bg) : -3.4e38f;
                s_w[b][j] = aj;
                mx = fmaxf(mx, aj);
            }
            mx = wave_max(mx);
            float se = 0.f;
            for (int j = lane; j < i; j += 32) {
                float raw = s_w[b][j];
                float e = (raw > -1e38f) ? __expf(raw - mx) : 0.f;
                s_w[b][j] = e;
                se += e;
            }
            se = wave_sum(se);
            const float inv = 1.f / se;

            float acc[10];
            #pragma unroll
            for (int t = 0; t < 10; ++t) acc[t] = 0.f;
            const float* bufb = Hnext + (size_t)(b0 + b) * NN * DH;
            for (int j = 0; j < i; ++j) {
                float wj = s_w[b][j] * inv;
                const float* row = bufb + (size_t)j * DH;
                #pragma unroll                 // unconditional: overreads stay
                for (int t = 0; t < 10; ++t)   // inside the batch's buffer and
                    acc[t] += wj * row[lane + 32 * t];  // are masked at store.
            }
            #pragma unroll
            for (int t = 0; t < 10; ++t) {
                int d = lane + 32 * t;
                if (d < DH) s_M[b][d] = acc[t];
            }
        }
        __syncthreads();

        // ---- phase 2: 4 x WMMA GEMM (16x300)@(300x900), A from LDS ----
        for (int job = wave; job < 4 * 57; job += 16) {
            const int g  = job / 57;
            const int n0 = (job - g * 57) * 16;
            const float* Amat = (g < 2) ? &s_x[0][0] : &s_M[0][0];
            const float* Wt; const float* bia;
            if      (g == 0) { Wt = cwih; bia = cbih; }   // gi_c = x @ cwih.T
            else if (g == 1) { Wt = pwhh; bia = pbhh; }   // gh_p = x @ pwhh.T
            else if (g == 2) { Wt = cwhh; bia = cbhh; }   // gh_c = M @ cwhh.T
            else             { Wt = pwih; bia = pbih; }   // gi_p = M @ pwih.T

            const int n = lane & 15, kh = lane >> 4;
            v8f acc = {0.f, 0.f, 0.f, 0.f, 0.f, 0.f, 0.f, 0.f};

            if (n0 + 16 <= 3 * DH) {          // full tile: guard-free hot loop
                const float* wbase = Wt + (size_t)(n0 + n) * DH + 2 * kh;
                __builtin_prefetch(wbase, 0, 1);
                for (int k0 = 0; k0 < DH; k0 += 4) {
                    v2f a  = a_frag_lds(Amat, k0, DH, lane);
                    v2f bf = *reinterpret_cast<const v2f*>(wbase + k0);
                    acc = wmma_f32(a, bf, acc);
                }
            } else {                          // single partial tile: clamp+mask
                const int col = n0 + n;
                const int cc  = col < 3 * DH ? col : 3 * DH - 1;
                const float msk = col < 3 * DH ? 1.f : 0.f;
                const float* wbase = Wt + (size_t)cc * DH + 2 * kh;
                for (int k0 = 0; k0 < DH; k0 += 4) {
                    v2f a  = a_frag_lds(Amat, k0, DH, lane);
                    v2f bf = *reinterpret_cast<const v2f*>(wbase + k0);
                    bf.x *= msk; bf.y *= msk;
                    acc = wmma_f32(a, bf, acc);
                }
            }

            const int col = n0 + n;
            if (col < 3 * DH) {
                const float bv = bia[col];
                float* gout = gws + (size_t)g * BQ * GPAD;
                #pragma unroll
                for (int r = 0; r < 8; ++r) {
                    int brow = r + kh * 8;
                    gout[(size_t)brow * GPAD + col] = acc[r] + bv;
                }
            }
        }
        __threadfence();
        __syncthreads();

        // ---- phase 3: GRU combine, write buf row / out slice ----
        for (int t = tid; t < BQ * DH; t += 512) {
            int b = t / DH, d = t - b * DH;
            const float* giC = gws + (size_t)(0 * BQ + b) * GPAD;
            const float* ghP = gws + (size_t)(1 * BQ + b) * GPAD;
            const float* ghC = gws + (size_t)(2 * BQ + b) * GPAD;
            const float* giP = gws + (size_t)(3 * BQ + b) * GPAD;

            float r  = sigmoidf(giC[d]        + ghC[d]);
            float z  = sigmoidf(giC[DH + d]   + ghC[DH + d]);
            float nn = tanhf   (giC[2*DH + d] + r * ghC[2*DH + d]);
            float C  = (1.f - z) * nn + z * s_M[b][d];

            float rp = sigmoidf(giP[d]        + ghP[d]);
            float zp = sigmoidf(giP[DH + d]   + ghP[DH + d]);
            float np = tanhf   (giP[2*DH + d] + rp * ghP[2*DH + d]);
            float P  = (1.f - zp) * np + zp * s_x[b][d];

            float v = C + P;
            Hnext[((size_t)(b0 + b) * NN + i) * DH + d] = v;
            out  [((size_t)(b0 + b) * NN + i) * ODIM + outoff + d] = v;
        }
        __threadfence();
        __syncthreads();

        // ---- phase 4: kdot[b][i] = buf[b,i,:] . wk ----
        {
            const int b = wave;
            const float* row = Hnext + ((size_t)(b0 + b) * NN + i) * DH;
            float kd = 0.f;
            for (int d = lane; d < DH; d += 32) kd += row[d] * wk[d];
            kd = wave_sum(kd);
            if (lane == 0) s_kdot[b][i] = kd;
        }
        __syncthreads();
    }
}

// ---------------------------------------------------------------------------
// Host launcher
// ---------------------------------------------------------------------------
extern "C" void kernel_launch(void* const* d_in, const int* in_sizes, int n_in,
                              void* d_out, int out_size, void* d_ws, size_t ws_size,
                              hipStream_t stream) {
    (void)in_sizes; (void)n_in; (void)out_size; (void)ws_size;

    const float* features = (const float*)d_in[0];
    const float* fc1_w    = (const float*)d_in[1];
    const float* fc1_b    = (const float*)d_in[2];
    const float* gat_w    = (const float*)d_in[3];
    const float* gat_b    = (const float*)d_in[4];
    const float* gc_wih   = (const float*)d_in[5];
    const float* gc_whh   = (const float*)d_in[6];
    const float* gc_bih   = (const float*)d_in[7];
    const float* gc_bhh   = (const float*)d_in[8];
    const float* gp_wih   = (const float*)d_in[9];
    const float* gp_whh   = (const float*)d_in[10];
    const float* gp_bih   = (const float*)d_in[11];
    const float* gp_bhh   = (const float*)d_in[12];
    const int*   adj      = (const int*)d_in[13];

    float* out = (float*)d_out;
    float* ws  = (float*)d_ws;

    const size_t hsz = (size_t)ROWS * DH;              // 2,457,600 floats
    float* Hb[3] = { ws, ws + hsz, ws + 2 * hsz };
    float* gws   = ws + 3 * hsz;                       // 4 * 4*16*912 floats

    fc1_kernel<<<dim3(64, 19), 256, 0, stream>>>(features, fc1_w, fc1_b, Hb[0], out);
    copy_feat_kernel<<<4096, 256, 0, stream>>>(features, out);

    for (int l = 0; l < LL; ++l) {
        dag_scan_kernel<<<4, 512, 0, stream>>>(
            Hb[l], Hb[l + 1],
            gc_wih + (size_t)l * 3 * DH * DH, gc_whh + (size_t)l * 3 * DH * DH,
            gc_bih + (size_t)l * 3 * DH,      gc_bhh + (size_t)l * 3 * DH,
            gp_wih + (size_t)l * 3 * DH * DH, gp_whh + (size_t)l * 3 * DH * DH,
            gp_bih + (size_t)l * 3 * DH,      gp_bhh + (size_t)l * 3 * DH,
            gat_w + (size_t)l * 2 * DH, gat_b + l,
            adj, gws, out, DH * (l + 1));
    }
}